// DynamicMemoryCell_22926535426539
// MI455X (gfx1250) — compile-verified
//
#include <hip/hip_runtime.h>
#include <hip/hip_bf16.h>

typedef float v2f __attribute__((ext_vector_type(2)));
typedef float v8f __attribute__((ext_vector_type(8)));

#define BATCH   8
#define SEQ     2048
#define IN_DIM  256
#define MEM_DIM 256
#define CTX_DIM 128
#define CAT_DIM (IN_DIM + MEM_DIM)      // 512
#define ROWS    (BATCH * SEQ)           // 16384

// One fp32 WMMA rank-4 update: D = A(16x4) * B(4x16) + C, full precision.
__device__ __forceinline__ v8f wmma4(v8f c, v2f a, v2f b) {
    return __builtin_amdgcn_wmma_f32_16x16x4_f32(false, a, false, b,
                                                 (short)0, c, false, false);
}

// ---------------------------------------------------------------------------
// Kernel 1: Q/K projections.  16 rows per block; 8 waves x 2 tiles each
// cover 8 Q n-tiles (CTX=128) + 8 K n-tiles.  K-reduction = 256 -> 64 WMMAs.
// ---------------------------------------------------------------------------
__global__ __launch_bounds__(256)
void proj_qk_kernel(const float* __restrict__ x, const float* __restrict__ m,
                    const float* __restrict__ Wq, const float* __restrict__ bq,
                    const float* __restrict__ Wk, const float* __restrict__ bk,
                    float* __restrict__ Qbuf, float* __restrict__ Kbuf) {
    const int rowbase = blockIdx.x * 16;
    const int lane = threadIdx.x & 31;
    const int wave = threadIdx.x >> 5;
    const int nn = lane & 15;        // A: M row / B: N col / D: col
    const int h  = lane >> 4;        // K-pair selector

    for (int tt = 0; tt < 2; ++tt) {
        const int t = wave * 2 + tt;           // 0..15
        const bool isQ = (t < 8);
        const int nbase = (isQ ? t : t - 8) * 16;
        const float* A    = isQ ? x   : m;
        const float* W    = isQ ? Wq  : Wk;
        const float* bias = isQ ? bq  : bk;
        float*       dst  = isQ ? Qbuf : Kbuf;

        const float* arow = A + (size_t)(rowbase + nn) * IN_DIM;
        const float* wrow = W + (size_t)(nbase + nn) * IN_DIM;

        v8f c = {};
#pragma unroll 8
        for (int s = 0; s < IN_DIM / 4; ++s) {
            const int d = 4 * s + 2 * h;
            c = wmma4(c, *(const v2f*)(arow + d), *(const v2f*)(wrow + d));
        }
        const float bb = bias[nbase + nn];
#pragma unroll
        for (int i = 0; i < 8; ++i) {
            const int M = i + 8 * h;
            dst[(size_t)(rowbase + M) * CTX_DIM + nbase + nn] = c[i] + bb;
        }
    }
}

// ---------------------------------------------------------------------------
// Kernel 2: attention scores + softmax, fused via 128KB LDS row-block.
// Block = (batch b, 16 query rows).  Wave w computes k-tiles w, w+8, ...
// (32 WMMAs each, CTX=128 reduction), deposits 16x16 score tiles into LDS,
// then the whole block does a cooperative softmax over the 16x2048 panel
// and streams probs to global (coalesced).
// ---------------------------------------------------------------------------
__global__ __launch_bounds__(256)
void attn_kernel(const float* __restrict__ Qbuf, const float* __restrict__ Kbuf,
                 float* __restrict__ probs) {
    extern __shared__ float smem[];          // 16 * 2048 fp32 = 128 KB
    __shared__ float sred[256];

    const int b     = blockIdx.x >> 7;       // 128 q-tiles per batch
    const int qbase = (blockIdx.x & 127) * 16;
    const int lane  = threadIdx.x & 31;
    const int wave  = threadIdx.x >> 5;
    const int nn    = lane & 15;
    const int h     = lane >> 4;

    // Preload this wave's Q A-operand (16x128 tile -> 64 regs/lane).
    const float* qrow = Qbuf + (size_t)(b * SEQ + qbase + nn) * CTX_DIM;
    float qa[64];
#pragma unroll
    for (int s = 0; s < 32; ++s) {
        const v2f v = *(const v2f*)(qrow + 4 * s + 2 * h);
        qa[2 * s] = v[0]; qa[2 * s + 1] = v[1];
    }

    for (int i = 0; i < 16; ++i) {
        const int kbase = (wave + 8 * i) * 16;
        const float* krow = Kbuf + (size_t)(b * SEQ + kbase + nn) * CTX_DIM;
        __builtin_prefetch(krow + CTX_DIM * 16, 0, 1);   // next tile hint
        v8f c = {};
#pragma unroll
        for (int s = 0; s < 32; ++s) {
            v2f a; a[0] = qa[2 * s]; a[1] = qa[2 * s + 1];
            c = wmma4(c, a, *(const v2f*)(krow + 4 * s + 2 * h));
        }
#pragma unroll
        for (int j = 0; j < 8; ++j)
            smem[(j + 8 * h) * SEQ + kbase + nn] = c[j];
    }
    __syncthreads();

    // Softmax: thread t owns row r = t>>4, columns (t&15) + 16*i.
    const int r  = threadIdx.x >> 4;
    const int jj = threadIdx.x & 15;

    float mx = -3.402823466e+38f;
    for (int col = jj; col < SEQ; col += 16)
        mx = fmaxf(mx, smem[r * SEQ + col]);
    sred[threadIdx.x] = mx;
    __syncthreads();
    for (int off = 8; off >= 1; off >>= 1) {
        if (jj < off)
            sred[threadIdx.x] = fmaxf(sred[threadIdx.x], sred[threadIdx.x + off]);
        __syncthreads();
    }
    mx = sred[r * 16];
    __syncthreads();

    float sum = 0.f;
    for (int col = jj; col < SEQ; col += 16) {
        const float e = __expf(smem[r * SEQ + col] - mx);
        smem[r * SEQ + col] = e;
        sum += e;
    }
    sred[threadIdx.x] = sum;
    __syncthreads();
    for (int off = 8; off >= 1; off >>= 1) {
        if (jj < off)
            sred[threadIdx.x] += sred[threadIdx.x + off];
        __syncthreads();
    }
    const float inv = 1.0f / sred[r * 16];

    float* orow = probs + ((size_t)b * SEQ + qbase + r) * SEQ;
    for (int col = jj; col < SEQ; col += 16)
        orow[col] = smem[r * SEQ + col] * inv;
}

// ---------------------------------------------------------------------------
// Kernel 3: fused GRU.  16 rows per block.  Phase 1: u and r gates (shared A
// loads, 2 WMMAs per K-step), stage sigmoid(u) and sigmoid(r)*m in LDS.
// Phase 2: candidate GEMM reads [x ; r*m] (second half from LDS), then
// new_mem = (1-u)*m + u*tanh(cand).  K-split at d=256 is step-uniform.
// ---------------------------------------------------------------------------
__global__ __launch_bounds__(256)
void gru_kernel(const float* __restrict__ x, const float* __restrict__ m,
                const float* __restrict__ Wu, const float* __restrict__ bu,
                const float* __restrict__ Wr, const float* __restrict__ br,
                const float* __restrict__ Wc, const float* __restrict__ bc,
                float* __restrict__ newmem) {
    __shared__ float uu[16][MEM_DIM];   // 16 KB
    __shared__ float rm[16][MEM_DIM];   // 16 KB

    const int rowbase = blockIdx.x * 16;
    const int lane = threadIdx.x & 31;
    const int wave = threadIdx.x >> 5;
    const int nn = lane & 15;
    const int h  = lane >> 4;

    const float* xrow = x + (size_t)(rowbase + nn) * IN_DIM;
    const float* mrow = m + (size_t)(rowbase + nn) * MEM_DIM;

    for (int tt = 0; tt < 2; ++tt) {
        const int nbase = (wave * 2 + tt) * 16;
        const float* wu = Wu + (size_t)(nbase + nn) * CAT_DIM;
        const float* wr = Wr + (size_t)(nbase + nn) * CAT_DIM;
        v8f cu = {}, cr = {};
#pragma unroll 8
        for (int s = 0; s < 64; ++s) {               // d in [0,256): x half
            const int d = 4 * s + 2 * h;
            const v2f a = *(const v2f*)(xrow + d);
            cu = wmma4(cu, a, *(const v2f*)(wu + d));
            cr = wmma4(cr, a, *(const v2f*)(wr + d));
        }
#pragma unroll 8
        for (int s = 64; s < 128; ++s) {             // d in [256,512): m half
            const int d = 4 * s + 2 * h;
            const v2f a = *(const v2f*)(mrow + d - IN_DIM);
            cu = wmma4(cu, a, *(const v2f*)(wu + d));
            cr = wmma4(cr, a, *(const v2f*)(wr + d));
        }
        const float bbu = bu[nbase + nn], bbr = br[nbase + nn];
#pragma unroll
        for (int i = 0; i < 8; ++i) {
            const int M = i + 8 * h;
            const float ug = 1.f / (1.f + __expf(-(cu[i] + bbu)));
            const float rg = 1.f / (1.f + __expf(-(cr[i] + bbr)));
            uu[M][nbase + nn] = ug;
            rm[M][nbase + nn] = rg * m[(size_t)(rowbase + M) * MEM_DIM + nbase + nn];
        }
    }
    __syncthreads();

    for (int tt = 0; tt < 2; ++tt) {
        const int nbase = (wave * 2 + tt) * 16;
        const float* wc = Wc + (size_t)(nbase + nn) * CAT_DIM;
        v8f cc = {};
#pragma unroll 8
        for (int s = 0; s < 64; ++s) {
            const int d = 4 * s + 2 * h;
            cc = wmma4(cc, *(const v2f*)(xrow + d), *(const v2f*)(wc + d));
        }
#pragma unroll 8
        for (int s = 64; s < 128; ++s) {             // A from LDS (r*m)
            const int d = 4 * s + 2 * h - IN_DIM;
            cc = wmma4(cc, *(const v2f*)(&rm[nn][d]),
                           *(const v2f*)(wc + d + IN_DIM));
        }
        const float bbc = bc[nbase + nn];
#pragma unroll
        for (int i = 0; i < 8; ++i) {
            const int M = i + 8 * h;
            const float cand = tanhf(cc[i] + bbc);
            const float mm = m[(size_t)(rowbase + M) * MEM_DIM + nbase + nn];
            const float ug = uu[M][nbase + nn];
            newmem[(size_t)(rowbase + M) * MEM_DIM + nbase + nn] =
                (1.f - ug) * mm + ug * cand;
        }
    }
}

// ---------------------------------------------------------------------------
extern "C" void kernel_launch(void* const* d_in, const int* in_sizes, int n_in,
                              void* d_out, int out_size, void* d_ws, size_t ws_size,
                              hipStream_t stream) {
    const float* x  = (const float*)d_in[0];
    const float* m  = (const float*)d_in[1];
    const float* Wq = (const float*)d_in[2];
    const float* bq = (const float*)d_in[3];
    const float* Wk = (const float*)d_in[4];
    const float* bk = (const float*)d_in[5];
    // d_in[6]/d_in[7] = Wv/bv: dead (context_vector unused in reference)
    const float* Wu = (const float*)d_in[8];
    const float* bu = (const float*)d_in[9];
    const float* Wr = (const float*)d_in[10];
    const float* br = (const float*)d_in[11];
    const float* Wc = (const float*)d_in[12];
    const float* bc = (const float*)d_in[13];

    float* out    = (float*)d_out;
    float* newmem = out;                                   // [8,2048,256]
    float* probs  = out + (size_t)ROWS * MEM_DIM;          // [8,2048,2048]

    float* Qbuf = (float*)d_ws;                            // 8 MB
    float* Kbuf = Qbuf + (size_t)ROWS * CTX_DIM;           // 8 MB

    const dim3 grid(ROWS / 16), blk(256);
    proj_qk_kernel<<<grid, blk, 0, stream>>>(x, m, Wq, bq, Wk, bk, Qbuf, Kbuf);
    attn_kernel<<<grid, blk, 16 * SEQ * sizeof(float), stream>>>(Qbuf, Kbuf, probs);
    gru_kernel<<<grid, blk, 0, stream>>>(x, m, Wu, bu, Wr, br, Wc, bc, newmem);
}